// MixedMSEPowerImbalance_69690139345431
// MI455X (gfx1250) — compile-verified
//
#include <hip/hip_runtime.h>
#include <hip/hip_bf16.h>

#define B_      256
#define NBUS    2000
#define TOTAL   (B_ * NBUS)        // 512000
#define E_      8388608
#define NY      (B_ * NBUS * 2)    // 1024000
#define DEG2RAD 0.017453292519943295f
#define INV_SN  0.01f

typedef float v2f __attribute__((ext_vector_type(2)));
typedef float v8f __attribute__((ext_vector_type(8)));
typedef float v4f __attribute__((ext_vector_type(4)));
typedef int   v4i __attribute__((ext_vector_type(4)));

// ---------------------------------------------------------------------------
// Wave-wide exact f32 sum using V_WMMA_F32_16X16X4_F32 with B = ones.
// A layout (16x4 f32): lanes 0-15 supply A[M][K=0] (vgpr0) / A[M][K=1] (vgpr1),
// lanes 16-31 supply A[M][K=2]/A[M][K=3].  With a = {v, 0}, row M of D is
// v(lane M) + v(lane M+16).  Each lane then sums its 8 D vgprs (rows 0-7 for
// lanes<16, rows 8-15 for lanes>=16); one xor-16 shuffle gives the full sum.
// EXEC must be all ones at the call site (whole-wave control flow only).
// ---------------------------------------------------------------------------
__device__ __forceinline__ float wave_sum_wmma(float v) {
    v2f a; a.x = v;    a.y = 0.0f;
    v2f b; b.x = 1.0f; b.y = 1.0f;
    v8f c = {};
    v8f d = __builtin_amdgcn_wmma_f32_16x16x4_f32(
        /*neg_a=*/false, a, /*neg_b=*/false, b,
        /*c_mod=*/(short)0, c, /*reuse_a=*/false, /*reuse_b=*/false);
    float s = d[0] + d[1] + d[2] + d[3] + d[4] + d[5] + d[6] + d[7];
    s += __shfl_xor(s, 16, 32);
    return s;
}

// Block-level reduce (blockDim.x multiple of 32, <= 1024) + one global atomic.
__device__ __forceinline__ void block_reduce_atomic(float v, float* target) {
    __shared__ float red[32];
    const int lane   = threadIdx.x & 31;
    const int wave   = threadIdx.x >> 5;
    const int nwaves = blockDim.x >> 5;
    float ws = wave_sum_wmma(v);
    __syncthreads();                    // protect red[] across repeated calls
    if (lane == 0) red[wave] = ws;
    __syncthreads();
    if (threadIdx.x < 32) {             // whole wave 0 => EXEC all ones
        float x  = (lane < nwaves) ? red[lane] : 0.0f;
        float bs = wave_sum_wmma(x);
        if (lane == 0) unsafeAtomicAdd(target, bs);
    }
}

// ---------------------------------------------------------------------------
// K1: per-node preprocessing: node[n] = { e, f, p, q }; zero agg + accums.
//     y index for node n=(bi,bus): bi*4000 + bus*2 == 2n.
// ---------------------------------------------------------------------------
__global__ void prep_kernel(const float* __restrict__ y_pred,
                            const float* __restrict__ y_std,
                            const float* __restrict__ y_mean,
                            const float* __restrict__ x_input,
                            const float* __restrict__ x_std,
                            const float* __restrict__ x_mean,
                            float4* __restrict__ node,
                            float2* __restrict__ agg,
                            float*  __restrict__ accums) {
    const int gid    = blockIdx.x * blockDim.x + threadIdx.x;
    const int stride = gridDim.x * blockDim.x;
    if (gid < 4) accums[gid] = 0.0f;
    for (int n = gid; n < TOTAL; n += stride) {
        const int bus = n % NBUS;
        const float vm = y_pred[2 * n]     * y_std[2 * bus]     + y_mean[2 * bus];
        const float va = (y_pred[2 * n + 1] * y_std[2 * bus + 1] + y_mean[2 * bus + 1]) * DEG2RAD;
        float sv, cv;
        __sincosf(va, &sv, &cv);        // hw v_sin/v_cos, |va| < ~1 rad here
        const float p = (x_input[7 * n]     * x_std[7 * bus]     + x_mean[7 * bus])     * INV_SN;
        const float q = (x_input[7 * n + 1] * x_std[7 * bus + 1] + x_mean[7 * bus + 1]) * INV_SN;
        node[n] = make_float4(vm * cv, vm * sv, p, q);
        agg[n]  = make_float2(0.0f, 0.0f);
    }
}

// ---------------------------------------------------------------------------
// K2: MSE partial sums (means cancel: diff = (pred-true)*y_std). float4 path.
// ---------------------------------------------------------------------------
__global__ void mse_kernel(const float4* __restrict__ yp,
                           const float4* __restrict__ yt,
                           const float*  __restrict__ y_std,
                           float* __restrict__ mse_acc) {
    const int gid    = blockIdx.x * blockDim.x + threadIdx.x;
    const int stride = gridDim.x * blockDim.x;
    const int nv     = NY / 4;
    float acc = 0.0f;
    for (int v = gid; v < nv; v += stride) {
        const float4 p = yp[v];
        const float4 t = yt[v];
        const int col4 = (v * 4) % (NBUS * 2);         // NBUS*2 % 4 == 0
        const float4 s = *(const float4*)(y_std + col4);
        float dx = (p.x - t.x) * s.x;
        float dy = (p.y - t.y) * s.y;
        float dz = (p.z - t.z) * s.z;
        float dw = (p.w - t.w) * s.w;
        acc += dx * dx + dy * dy + dz * dz + dw * dw;
    }
    block_reduce_atomic(acc, mse_acc);
}

// ---------------------------------------------------------------------------
// K3: edge kernel — gather node (e,f) at both endpoints, scatter P/Q onto i.
// ---------------------------------------------------------------------------
__device__ __forceinline__ void do_edge(int i, int j, float ra, float xa,
                                        float em0, float es0, float em1, float es1,
                                        const float4* __restrict__ node,
                                        float* __restrict__ agg) {
    const float r = ra * es0 + em0;
    const float x = xa * es1 + em1;
    const float inv = 1.0f / (r * r + x * x);
    const float g =  r * inv;
    const float b = -x * inv;
    const float4 ni = node[i];
    const float4 nj = node[j];
    const float t1 = ni.x * nj.x - ni.x * ni.x + ni.y * nj.y - ni.y * ni.y;
    const float t2 = ni.y * nj.x - ni.x * nj.y;
    const float P = g * t1 + b * t2;
    const float Q = g * t2 - b * t1;
    unsafeAtomicAdd(agg + 2 * i,     P);   // global_atomic_add_f32, hits L2
    unsafeAtomicAdd(agg + 2 * i + 1, Q);
}

__global__ void edge_kernel(const int*   __restrict__ edge_index,
                            const float* __restrict__ edge_attr,
                            const float* __restrict__ edge_mean,
                            const float* __restrict__ edge_std,
                            const float4* __restrict__ node,
                            float* __restrict__ agg) {
    const int gid    = blockIdx.x * blockDim.x + threadIdx.x;
    const int stride = gridDim.x * blockDim.x;
    const float em0 = edge_mean[0], em1 = edge_mean[1];
    const float es0 = edge_std[0],  es1 = edge_std[1];
    const v4i* ip = (const v4i*)edge_index;
    const v4i* jp = (const v4i*)(edge_index + E_);
    const v4f* ap = (const v4f*)edge_attr;
    const int nv = E_ / 4;
    for (int t = gid; t < nv; t += stride) {
        // one-shot streams: non-temporal so the hot node/agg set stays in L2
        const v4i ii = __builtin_nontemporal_load(ip + t);
        const v4i jj = __builtin_nontemporal_load(jp + t);
        const v4f a0 = __builtin_nontemporal_load(ap + 2 * t);
        const v4f a1 = __builtin_nontemporal_load(ap + 2 * t + 1);
        do_edge(ii.x, jj.x, a0.x, a0.y, em0, es0, em1, es1, node, agg);
        do_edge(ii.y, jj.y, a0.z, a0.w, em0, es0, em1, es1, node, agg);
        do_edge(ii.z, jj.z, a1.x, a1.y, em0, es0, em1, es1, node, agg);
        do_edge(ii.w, jj.w, a1.z, a1.w, em0, es0, em1, es1, node, agg);
    }
}

// ---------------------------------------------------------------------------
// K4: per-node imbalance: dP = p - agg.x, dQ = q - agg.y; masked sums.
// ---------------------------------------------------------------------------
__global__ void phys_kernel(const float4* __restrict__ node,
                            const float2* __restrict__ agg,
                            const unsigned char* __restrict__ mask,
                            float* __restrict__ accums) {
    const int gid    = blockIdx.x * blockDim.x + threadIdx.x;
    const int stride = gridDim.x * blockDim.x;
    float accS = 0.0f, accM = 0.0f;
    for (int n = gid; n < TOTAL; n += stride) {
        const float4 nd = node[n];
        const float2 ag = agg[n];
        const float dP = nd.z - ag.x;
        const float dQ = nd.w - ag.y;
        const float m  = mask[n % NBUS] ? 1.0f : 0.0f;
        accS += (dP * dP + dQ * dQ) * m;
        accM += m;
    }
    block_reduce_atomic(accS, accums + 1);
    block_reduce_atomic(accM, accums + 2);
}

// K5: combine scalar.
__global__ void finalize_kernel(const float* __restrict__ accums,
                                float* __restrict__ out) {
    const float mse  = accums[0] / (float)NY;
    const float phys = accums[1] / accums[2];
    out[0] = 0.9f * mse + 0.1f * 0.02f * phys;
}

extern "C" void kernel_launch(void* const* d_in, const int* in_sizes, int n_in,
                              void* d_out, int out_size, void* d_ws, size_t ws_size,
                              hipStream_t stream) {
    (void)in_sizes; (void)n_in; (void)out_size; (void)ws_size;
    const float* y_pred    = (const float*)d_in[0];
    const float* y_true    = (const float*)d_in[1];
    const float* x_input   = (const float*)d_in[2];
    const float* edge_attr = (const float*)d_in[3];
    const float* x_mean    = (const float*)d_in[4];
    const float* x_std     = (const float*)d_in[5];
    const float* y_mean    = (const float*)d_in[6];
    const float* y_std     = (const float*)d_in[7];
    const float* edge_mean = (const float*)d_in[8];
    const float* edge_std  = (const float*)d_in[9];
    const int*   edge_idx  = (const int*)d_in[10];
    const unsigned char* mask = (const unsigned char*)d_in[11];
    float* out = (float*)d_out;

    // workspace layout: node table (8 MB) | agg (4 MB) | accums (16 B)
    float4* node   = (float4*)d_ws;
    float*  agg    = (float*)((char*)d_ws + (size_t)TOTAL * 16);
    float*  accums = (float*)((char*)d_ws + (size_t)TOTAL * 16 + (size_t)TOTAL * 8);

    const int TPB = 256;  // 8 waves (wave32)
    prep_kernel<<<TOTAL / TPB, TPB, 0, stream>>>(y_pred, y_std, y_mean,
                                                 x_input, x_std, x_mean,
                                                 node, (float2*)agg, accums);
    mse_kernel<<<(NY / 4) / TPB, TPB, 0, stream>>>((const float4*)y_pred,
                                                   (const float4*)y_true,
                                                   y_std, accums);
    edge_kernel<<<(E_ / 4) / TPB, TPB, 0, stream>>>(edge_idx, edge_attr,
                                                    edge_mean, edge_std,
                                                    node, agg);
    phys_kernel<<<TOTAL / TPB, TPB, 0, stream>>>(node, (const float2*)agg,
                                                 mask, accums);
    finalize_kernel<<<1, 1, 0, stream>>>(accums, out);
}